// FrequencyDomainRegularizer_19052474925671
// MI455X (gfx1250) — compile-verified
//
#include <hip/hip_runtime.h>
#include <hip/hip_bf16.h>
#include <math.h>

// ---------------------------------------------------------------------------
// FrequencyDomainRegularizer for MI455X (gfx1250, wave32, WMMA).
// DFT-as-GEMM using V_WMMA_F32_16X16X4_F32 (exact f32 tensor path).
// ---------------------------------------------------------------------------

typedef __attribute__((ext_vector_type(2))) float v2f;
typedef __attribute__((ext_vector_type(8))) float v8f;

#define BATCH 128
#define HDIM 512
#define WDIM 512
#define WR 257            // rfft width = W/2 + 1
#define KCOLS 272         // 17 tiles of 16 columns (k > 256 masked out)
#define KBLOCKS 17
#define NBINS 32
#define EPSV 1e-8f
#define F_MAXV 0.70710678118654752440f

// workspace layout (in floats)
#define OFF_CW   0                         // 512 x 272  cos(2*pi*w*k/512)
#define OFF_SWN  (512*272)                 // 512 x 272  -sin(...)
#define OFF_CH   (2*512*272)               // 512 x 512  cos(2*pi*ky*h/512)
#define OFF_SHP  (2*512*272 + 512*512)     // 512 x 512  +sin
#define OFF_SHN  (2*512*272 + 2*512*512)   // 512 x 512  -sin
#define OFF_PART (2*512*272 + 3*512*512)   // 2176 x 32 per-block bin partials
// total floats = 1,134,592  (~4.6 MB)

static __device__ __forceinline__ v8f wmma4(v2f a, v2f b, v8f c) {
  // D = A(16x4, f32) * B(4x16, f32) + C(16x16, f32)
  return __builtin_amdgcn_wmma_f32_16x16x4_f32(
      /*neg_a=*/false, a, /*neg_b=*/false, b,
      /*c_mod=*/(short)0, c, /*reuse_a=*/false, /*reuse_b=*/false);
}

// Emulate jnp.searchsorted(linspace(0,f_max,33), f, 'right') - 1, with
// out-of-range (f == f_max corner) returning -1 (discard).
static __device__ __forceinline__ int radial_bin(int ky, int k) {
  float fyv = (ky < 256) ? (float)ky : (float)(ky - 512);
  float fy = fyv * (1.0f / 512.0f);
  float fx = (float)k * (1.0f / 512.0f);
  float f = sqrtf(fy * fy + fx * fx);
  const float STEP = F_MAXV / 32.0f;
  int idx = (int)floorf(f / STEP);
  if (idx > 32) idx = 32;
  if (idx < 0) idx = 0;
  while (idx < 32 && (float)(idx + 1) * STEP <= f) ++idx;
  while (idx > 0 && (float)idx * STEP > f) --idx;
  if (idx >= NBINS) return -1;
  return idx;
}

// ---------------------------------------------------------------------------
// Kernel 1: build DFT twiddle tables (phase reduced mod 512 for f32 accuracy)
// ---------------------------------------------------------------------------
__global__ __launch_bounds__(256) void fdr_build_tables(float* __restrict__ ws) {
  const int i = blockIdx.x * blockDim.x + threadIdx.x;
  const float TPO512 = 6.28318530717958647692f / 512.0f;
  if (i < 512 * KCOLS) {
    int w = i / KCOLS, k = i % KCOLS;
    int p = (w * k) & 511;
    float th = (float)p * TPO512;
    ws[OFF_CW + i] = cosf(th);
    ws[OFF_SWN + i] = -sinf(th);   // rfft: Xi = -sum x*sin
  }
  if (i < 512 * 512) {
    int ky = i >> 9, h = i & 511;
    int p = (ky * h) & 511;
    float th = (float)p * TPO512;
    ws[OFF_CH + i] = cosf(th);
    ws[OFF_SHP + i] = sinf(th);    // Zr = C*Yr + S*Yi
    ws[OFF_SHN + i] = -sinf(th);   // Zi = C*Yi - S*Yr
  }
}

// ---------------------------------------------------------------------------
// Kernel 2: per (image, 16-column block): row-rFFT tile -> LDS, column FFT,
// power spectrum, radial binning, deterministic block reduction -> partials.
// ---------------------------------------------------------------------------
__global__ __launch_bounds__(256) void fdr_main(const float* __restrict__ noisy,
                                                const float* __restrict__ denoised,
                                                const float* __restrict__ ws,
                                                float* __restrict__ partial) {
  __shared__ float lds[16384];   // 64 KB: Yr = [0,8192), Yi = [8192,16384)

  const int bid = blockIdx.x;
  const int b  = bid / KBLOCKS;       // b-major: consecutive blocks share image (L2)
  const int kb = bid % KBLOCKS;
  const int tid  = threadIdx.x;
  const int lane = tid & 31;
  const int wave = tid >> 5;
  const int lm = lane & 15;           // M / N index within tile
  const int lh = lane >> 4;           // K half-select (0 or 1)

  const float* Cw  = ws + OFF_CW;
  const float* Swn = ws + OFF_SWN;
  const float* Ch  = ws + OFF_CH;
  const float* Shp = ws + OFF_SHP;
  const float* Shn = ws + OFF_SHN;

  const int kcol = kb * 16 + lm;      // output frequency column (0..271)
  const size_t imgBase = (size_t)b * (HDIM * WDIM);

  // ---- Stage 1: Yr/Yi[h, kcol] = sum_w R[h,w] * {cos,-sin}(2*pi*w*k/512) ----
  for (int t = 0; t < 4; ++t) {
    const int htile = wave * 4 + t;            // 8 waves x 4 = 32 tiles of 16 rows
    const int m = htile * 16 + lm;
    v8f cr = {};
    v8f ci = {};
    const float* nrow = noisy + imgBase + (size_t)m * WDIM;
    const float* drow = denoised + imgBase + (size_t)m * WDIM;
    for (int kk = 0; kk < 512; kk += 4) {
      const int w0 = kk + 2 * lh;              // A lane layout: K = 2*(lane/16)+v
      float2 nz = *reinterpret_cast<const float2*>(nrow + w0);
      float2 dn = *reinterpret_cast<const float2*>(drow + w0);
      v2f a;  a.x = nz.x - dn.x;  a.y = nz.y - dn.y;
      v2f bc, bs;                               // B lane layout: K = 2*(lane/16)+v, N = lm
      bc.x = Cw[(size_t)w0 * KCOLS + kcol];
      bc.y = Cw[(size_t)(w0 + 1) * KCOLS + kcol];
      bs.x = Swn[(size_t)w0 * KCOLS + kcol];
      bs.y = Swn[(size_t)(w0 + 1) * KCOLS + kcol];
      cr = wmma4(a, bc, cr);
      ci = wmma4(a, bs, ci);
    }
    // C/D layout: M = v + 8*(lane/16), N = lm
    for (int v = 0; v < 8; ++v) {
      const int row = htile * 16 + v + 8 * lh;
      lds[row * 16 + lm] = cr[v];
      lds[8192 + row * 16 + lm] = ci[v];
    }
  }
  __syncthreads();

  // ---- Stage 2: Z[ky, kcol] = sum_h e^{-i 2 pi h ky/512} Y[h, kcol] ----
  float priv[NBINS];
  for (int i = 0; i < NBINS; ++i) priv[i] = 0.0f;

  for (int t = 0; t < 4; ++t) {
    const int kytile = wave * 4 + t;
    const int mrow = kytile * 16 + lm;
    v8f czr = {};
    v8f czi = {};
    const float* chrow = Ch + (size_t)mrow * 512;
    const float* sprow = Shp + (size_t)mrow * 512;
    const float* snrow = Shn + (size_t)mrow * 512;
    for (int kk = 0; kk < 512; kk += 4) {
      const int h0 = kk + 2 * lh;
      float2 c2 = *reinterpret_cast<const float2*>(chrow + h0);
      float2 p2 = *reinterpret_cast<const float2*>(sprow + h0);
      float2 n2 = *reinterpret_cast<const float2*>(snrow + h0);
      v2f ac;  ac.x = c2.x;  ac.y = c2.y;
      v2f ap;  ap.x = p2.x;  ap.y = p2.y;
      v2f an;  an.x = n2.x;  an.y = n2.y;
      v2f br, bi;
      br.x = lds[h0 * 16 + lm];
      br.y = lds[(h0 + 1) * 16 + lm];
      bi.x = lds[8192 + h0 * 16 + lm];
      bi.y = lds[8192 + (h0 + 1) * 16 + lm];
      czr = wmma4(ac, br, czr);   // + cos*Yr
      czr = wmma4(ap, bi, czr);   // + sin*Yi
      czi = wmma4(ac, bi, czi);   // + cos*Yi
      czi = wmma4(an, br, czi);   // - sin*Yr
    }
    if (kcol <= 256) {
      for (int v = 0; v < 8; ++v) {
        const int ky = kytile * 16 + v + 8 * lh;
        // ortho norm: |Z|^2 / (H*W)
        const float p = (czr[v] * czr[v] + czi[v] * czi[v]) * (1.0f / 262144.0f);
        const int bin = radial_bin(ky, kcol);
        if (bin >= 0) priv[bin] += p;
      }
    }
  }
  __syncthreads();

  // ---- Deterministic fixed-order block reduction (reuse LDS) ----
  for (int i = 0; i < NBINS; ++i) lds[tid * NBINS + i] = priv[i];
  __syncthreads();
  if (tid < NBINS) {
    float s = 0.0f;
    for (int t = 0; t < 256; ++t) s += lds[t * NBINS + tid];
    partial[(size_t)bid * NBINS + tid] = s;
  }
}

// ---------------------------------------------------------------------------
// Kernel 3: counts (geometry only), per-image PSD normalize, reference PSD,
// log-L1, scalar mean. One block, deterministic.
// ---------------------------------------------------------------------------
__global__ __launch_bounds__(256) void fdr_final(const float* __restrict__ noise_level,
                                                 const float* __restrict__ ws,
                                                 float* __restrict__ out) {
  __shared__ unsigned scnt[NBINS];
  __shared__ float fcnt[NBINS];
  __shared__ float sred[256];
  const int tid = threadIdx.x;
  if (tid < NBINS) scnt[tid] = 0u;
  __syncthreads();

  for (int i = tid; i < HDIM * WR; i += 256) {
    const int ky = i / WR, k = i % WR;
    const int bin = radial_bin(ky, k);
    if (bin >= 0) atomicAdd(&scnt[bin], 1u);   // integer: order-independent
  }
  __syncthreads();
  if (tid < NBINS) fcnt[tid] = (float)scnt[tid];
  __syncthreads();

  const float* partial = ws + OFF_PART;
  float acc = 0.0f;
  if (tid < BATCH) {
    const int b = tid;
    float psd[NBINS];
    float tot = 0.0f;
    for (int bin = 0; bin < NBINS; ++bin) {
      float s = 0.0f;
      for (int kb = 0; kb < KBLOCKS; ++kb)     // fixed order -> deterministic
        s += partial[((size_t)(b * KBLOCKS + kb)) * NBINS + bin];
      const float c = fcnt[bin];
      const float v = (c > 0.0f) ? (s / fmaxf(c, 1.0f)) : 0.0f;
      psd[bin] = v;
      tot += v;
    }
    tot = fmaxf(tot, EPSV);

    float refv[NBINS];
    float rtot = 0.0f;
    const float nl = noise_level[b];
    for (int i = 0; i < NBINS; ++i) {
      const float fb = (float)i * (F_MAXV / 31.0f);   // linspace(0, f_max, 32)
      const float shot = expf(-78.95683520871486f * fb * fb);  // -2*pi^2*sigma^2
      const float r = nl * shot + (1.0f - 0.5f * nl);
      refv[i] = r;
      rtot += r;
    }
    rtot = fmaxf(rtot, EPSV);

    for (int i = 0; i < NBINS; ++i) {
      const float lr = logf(psd[i] / tot + EPSV);
      const float lf = logf(refv[i] / rtot + EPSV);
      acc += fabsf(lr - lf);
    }
  }
  sred[tid] = acc;
  __syncthreads();
  if (tid == 0) {
    float s = 0.0f;
    for (int t = 0; t < 256; ++t) s += sred[t];
    out[0] = 0.05f * s / (float)(BATCH * NBINS);
  }
}

// ---------------------------------------------------------------------------
extern "C" void kernel_launch(void* const* d_in, const int* in_sizes, int n_in,
                              void* d_out, int out_size, void* d_ws, size_t ws_size,
                              hipStream_t stream) {
  (void)in_sizes; (void)n_in; (void)out_size; (void)ws_size;
  const float* noisy       = (const float*)d_in[0];
  const float* denoised    = (const float*)d_in[1];
  const float* noise_level = (const float*)d_in[2];
  float* ws  = (float*)d_ws;          // needs ~4.6 MB
  float* out = (float*)d_out;

  // 1) twiddle tables (recomputed every call: stateless/deterministic)
  fdr_build_tables<<<dim3((512 * 512 + 255) / 256), dim3(256), 0, stream>>>(ws);

  // 2) main DFT-as-WMMA pass: 128 images x 17 column blocks
  fdr_main<<<dim3(BATCH * KBLOCKS), dim3(256), 0, stream>>>(
      noisy, denoised, ws, ws + OFF_PART);

  // 3) scalar loss
  fdr_final<<<dim3(1), dim3(256), 0, stream>>>(noise_level, ws, out);
}